// MultiHeadSelfAttention_5772436046008
// MI455X (gfx1250) — compile-verified
//
#include <hip/hip_runtime.h>

typedef float v2f __attribute__((ext_vector_type(2)));
typedef float v8f __attribute__((ext_vector_type(8)));

constexpr int  Bc  = 4;
constexpr int  Sc  = 1024;
constexpr int  Dc  = 1024;
constexpr int  Hc  = 16;
constexpr int  DKc = 64;
constexpr int  DVc = 32;
constexpr float INV_SCALE = 0.125f;  // 1/sqrt(DK)

#define ZERO8 {0.f, 0.f, 0.f, 0.f, 0.f, 0.f, 0.f, 0.f}

// LDS byte offset of a pointer into a __shared__ array (AS3 offset for the
// VDST VGPR of global_load_async_to_lds_*).
__device__ __forceinline__ unsigned lds_offset(const void* p) {
  return (unsigned)(unsigned long long)(__attribute__((address_space(3))) const char*)p;
}

// Async copy 16B per lane: global address gaddr to LDS offset lds_off (ASYNCcnt-tracked).
__device__ __forceinline__ void async_copy_b128(unsigned lds_off, const float* gaddr) {
  asm volatile("global_load_async_to_lds_b128 %0, %1, off"
               :: "v"(lds_off), "v"(gaddr) : "memory");
}
__device__ __forceinline__ void wait_asynccnt0() {
  asm volatile("s_wait_asynccnt 0" ::: "memory");
}

__device__ __forceinline__ v8f wmma4(v2f a, v2f b, v8f acc) {
  return __builtin_amdgcn_wmma_f32_16x16x4_f32(false, a, false, b,
                                               (short)0, acc, false, false);
}

// ---------------------------------------------------------------------------
// Kernel 1: fused Q/K/V projections.  One block per (b,h, 16-row tile).
// The 16x1024 x-row tile (64KB, contiguous) is staged into LDS with async
// b128 loads; 10 waves then each compute one 16x16 output tile
// (4 Q + 4 K + 2 V column tiles), A-fragments via ds_load, W via L2.
// ---------------------------------------------------------------------------
__global__ __launch_bounds__(320) void qkv_kernel(
    const float* __restrict__ x,
    const float* __restrict__ Wq, const float* __restrict__ bq,
    const float* __restrict__ Wk, const float* __restrict__ bk,
    const float* __restrict__ Wv, const float* __restrict__ bv,
    float* __restrict__ qb, float* __restrict__ kb, float* __restrict__ vb) {
  __shared__ float xt[16 * 1024];  // 64KB row tile of x

  const int bh  = blockIdx.x >> 6;
  const int rt  = blockIdx.x & 63;
  const int b   = bh >> 4;
  const int h   = bh & 15;
  const int tid = threadIdx.x;

  // ---- async stage: 16384 contiguous floats -> LDS ----
  const float* gsrc = x + (size_t)b * Sc * Dc + (size_t)rt * 16 * Dc;
  const unsigned lds0 = lds_offset(xt);
  if (tid < 256) {
#pragma unroll
    for (int i = 0; i < 16; ++i) {
      const int f0 = (i * 256 + tid) * 4;
      async_copy_b128(lds0 + f0 * 4, gsrc + f0);
    }
  }
  wait_asynccnt0();
  __syncthreads();

  // ---- compute: wave c handles one 16x16 column tile ----
  const int c    = tid >> 5;       // 0..9
  const int lane = tid & 31;
  const int lo   = lane & 15;
  const int hi   = lane >> 4;
  const int kh   = hi * 2;

  const float* W;
  const float* bias;
  float*       out;
  int N, n0;
  if (c < 4) {
    N = DKc; n0 = c * 16;
    W = Wq + (size_t)h * Dc * DKc;  bias = bq + h * DKc;
    out = qb + (size_t)bh * Sc * DKc;
  } else if (c < 8) {
    N = DKc; n0 = (c - 4) * 16;
    W = Wk + (size_t)h * Dc * DKc;  bias = bk + h * DKc;
    out = kb + (size_t)bh * Sc * DKc;
  } else {
    N = DVc; n0 = (c - 8) * 16;
    W = Wv + (size_t)h * Dc * DVc;  bias = bv + h * DVc;
    out = vb + (size_t)bh * Sc * DVc;
  }

  const float* aL = xt + lo * Dc + kh;         // LDS reads (ds_load)
  const float* bP = W + (size_t)kh * N + n0 + lo;

  v8f acc = ZERO8;
#pragma unroll 8
  for (int k = 0; k < Dc; k += 4) {
    v2f a, bb;
    a.x  = aL[k];
    a.y  = aL[k + 1];
    bb.x = bP[k * N];
    bb.y = bP[(k + 1) * N];
    acc = wmma4(a, bb, acc);
  }

  const float bias_v = bias[n0 + lo];
#pragma unroll
  for (int j = 0; j < 8; ++j)
    out[(size_t)(rt * 16 + j + 8 * hi) * N + n0 + lo] = acc[j] + bias_v;
}

// ---------------------------------------------------------------------------
// Kernel 2: scores = q @ k^T / sqrt(DK).  One block per (b,h, 16-row tile);
// each wave holds the full 16x64 q strip in registers (32 VGPRs) and reuses
// it across 8 key column tiles -> 16 WMMAs per tile fed by 16 b64 loads.
// ---------------------------------------------------------------------------
__global__ __launch_bounds__(256) void scores_kernel(
    const float* __restrict__ qb, const float* __restrict__ kb,
    float* __restrict__ wts) {
  const int bh   = blockIdx.x >> 6;
  const int rt   = blockIdx.x & 63;
  const int tid  = threadIdx.x;
  const int wave = tid >> 5;
  const int lane = tid & 31;
  const int lo   = lane & 15;
  const int hi   = lane >> 4;
  const int kh   = hi * 2;

  const float* qrow = qb + (size_t)bh * Sc * DKc + (size_t)(rt * 16 + lo) * DKc + kh;
  v2f qa[16];
#pragma unroll
  for (int i = 0; i < 16; ++i) {
    qa[i].x = qrow[4 * i];
    qa[i].y = qrow[4 * i + 1];
  }

  const float* kbase = kb + (size_t)bh * Sc * DKc;
  float*       outb  = wts + (size_t)bh * Sc * Sc;

  for (int cti = 0; cti < 8; ++cti) {
    const int ct = wave * 8 + cti;
    const float* krow = kbase + (size_t)(ct * 16 + lo) * DKc + kh;
    if (cti < 7)  // prefetch next key tile rows into L2/L0
      __builtin_prefetch(krow + 16 * DKc, 0, 3);

    v8f acc = ZERO8;
#pragma unroll
    for (int i = 0; i < 16; ++i) {
      v2f kf;
      kf.x = krow[4 * i];
      kf.y = krow[4 * i + 1];
      acc = wmma4(qa[i], kf, acc);
    }
#pragma unroll
    for (int j = 0; j < 8; ++j)
      outb[(size_t)(rt * 16 + j + 8 * hi) * Sc + ct * 16 + lo] = acc[j] * INV_SCALE;
  }
}

// ---------------------------------------------------------------------------
// Kernel 3: in-place row softmax over S=1024 (one 256-thread block per row).
// ---------------------------------------------------------------------------
__global__ __launch_bounds__(256) void softmax_kernel(float* __restrict__ wts) {
  float* w = wts + (size_t)blockIdx.x * Sc;
  const int tid = threadIdx.x;
  __shared__ float red[256];

  float v0 = w[tid], v1 = w[tid + 256], v2 = w[tid + 512], v3 = w[tid + 768];
  float m = fmaxf(fmaxf(v0, v1), fmaxf(v2, v3));
  red[tid] = m;
  __syncthreads();
#pragma unroll
  for (int s = 128; s > 0; s >>= 1) {
    if (tid < s) red[tid] = fmaxf(red[tid], red[tid + s]);
    __syncthreads();
  }
  m = red[0];
  __syncthreads();

  float e0 = __expf(v0 - m), e1 = __expf(v1 - m);
  float e2 = __expf(v2 - m), e3 = __expf(v3 - m);
  red[tid] = (e0 + e1) + (e2 + e3);
  __syncthreads();
#pragma unroll
  for (int s = 128; s > 0; s >>= 1) {
    if (tid < s) red[tid] += red[tid + s];
    __syncthreads();
  }
  const float inv = 1.0f / red[0];

  w[tid]       = e0 * inv;
  w[tid + 256] = e1 * inv;
  w[tid + 512] = e2 * inv;
  w[tid + 768] = e3 * inv;
}

// ---------------------------------------------------------------------------
// Kernel 4: out_h = weights @ v in concat order [B,S,H*DV].
// One block per (b,h, 8 row tiles).  v[bh] (1024x32) is async-staged into
// LDS in two 64KB halves; each wave computes a 16x32 strip (2 accumulators
// sharing one A-fragment per k-step).
// ---------------------------------------------------------------------------
__global__ __launch_bounds__(256) void av_kernel(
    const float* __restrict__ wts, const float* __restrict__ vb,
    float* __restrict__ cbuf) {
  __shared__ float vt[512 * DVc];  // 64KB half of v[bh]

  const int bh   = blockIdx.x >> 3;
  const int rtg  = blockIdx.x & 7;
  const int b    = bh >> 4;
  const int h    = bh & 15;
  const int tid  = threadIdx.x;
  const int wave = tid >> 5;
  const int lane = tid & 31;
  const int lo   = lane & 15;
  const int hi   = lane >> 4;
  const int kh   = hi * 2;
  const int rt   = rtg * 8 + wave;

  const float* aP = wts + (size_t)bh * Sc * Sc + (size_t)(rt * 16 + lo) * Sc + kh;
  const unsigned lds0 = lds_offset(vt);
  const float* bL = vt + kh * DVc + lo;

  v8f acc0 = ZERO8, acc1 = ZERO8;
  for (int chunk = 0; chunk < 2; ++chunk) {
    if (chunk) __syncthreads();  // previous half fully consumed
    const float* gsrc = vb + (size_t)bh * Sc * DVc + chunk * 512 * DVc;
#pragma unroll
    for (int i = 0; i < 16; ++i) {
      const int f0 = (i * 256 + tid) * 4;
      async_copy_b128(lds0 + f0 * 4, gsrc + f0);
    }
    wait_asynccnt0();
    __syncthreads();

    const float* aC = aP + chunk * 512;
#pragma unroll 4
    for (int k = 0; k < 512; k += 4) {
      v2f a, b0, b1;
      a.x  = aC[k];
      a.y  = aC[k + 1];
      b0.x = bL[k * DVc];
      b0.y = bL[(k + 1) * DVc];
      b1.x = bL[k * DVc + 16];
      b1.y = bL[(k + 1) * DVc + 16];
      acc0 = wmma4(a, b0, acc0);
      acc1 = wmma4(a, b1, acc1);
    }
  }

#pragma unroll
  for (int j = 0; j < 8; ++j) {
    const int m = rt * 16 + j + 8 * hi;
    float* row = cbuf + ((size_t)(b * Sc + m) * Hc + h) * DVc;
    row[lo]      = acc0[j];
    row[16 + lo] = acc1[j];
  }
}

// ---------------------------------------------------------------------------
// Kernel 5: output = concat @ Wo + bo.  [4096 x 512] @ [512 x 1024].
// Each wave computes a 16x64 strip: 4 accumulators share one A-fragment.
// ---------------------------------------------------------------------------
__global__ __launch_bounds__(256) void oproj_kernel(
    const float* __restrict__ cbuf, const float* __restrict__ Wo,
    const float* __restrict__ bo, float* __restrict__ out) {
  const int t    = blockIdx.x * 8 + (threadIdx.x >> 5);  // 0..4095
  const int rt   = t >> 4;   // 0..255
  const int ctg  = t & 15;   // group of 4 column tiles
  const int lane = threadIdx.x & 31;
  const int lo   = lane & 15;
  const int hi   = lane >> 4;
  const int kh   = hi * 2;
  const int Kd   = Hc * DVc;  // 512

  const float* aP = cbuf + (size_t)(rt * 16 + lo) * Kd + kh;
  const float* bP = Wo + (size_t)kh * Dc + ctg * 64 + lo;

  v8f accs[4] = {ZERO8, ZERO8, ZERO8, ZERO8};
#pragma unroll 2
  for (int k = 0; k < Kd; k += 4) {
    v2f a;
    a.x = aP[k];
    a.y = aP[k + 1];
#pragma unroll
    for (int t4 = 0; t4 < 4; ++t4) {
      v2f bb;
      bb.x = bP[k * Dc + t4 * 16];
      bb.y = bP[(k + 1) * Dc + t4 * 16];
      accs[t4] = wmma4(a, bb, accs[t4]);
    }
  }

#pragma unroll
  for (int t4 = 0; t4 < 4; ++t4) {
    const int n0 = ctg * 64 + t4 * 16;
    const float bias_v = bo[n0 + lo];
#pragma unroll
    for (int j = 0; j < 8; ++j)
      out[(size_t)(rt * 16 + j + 8 * hi) * Dc + n0 + lo] = accs[t4][j] + bias_v;
  }
}

// ---------------------------------------------------------------------------
extern "C" void kernel_launch(void* const* d_in, const int* in_sizes, int n_in,
                              void* d_out, int out_size, void* d_ws, size_t ws_size,
                              hipStream_t stream) {
  const float* x  = (const float*)d_in[0];
  const float* Wq = (const float*)d_in[1];
  const float* bq = (const float*)d_in[2];
  const float* Wk = (const float*)d_in[3];
  const float* bk = (const float*)d_in[4];
  const float* Wv = (const float*)d_in[5];
  const float* bv = (const float*)d_in[6];
  const float* Wo = (const float*)d_in[7];
  const float* bo = (const float*)d_in[8];

  float* out = (float*)d_out;                    // [B,S,D]
  float* wts = out + (size_t)Bc * Sc * Dc;       // [B,H,S,S]

  float* qb   = (float*)d_ws;                    // [B*H, S, DK]
  float* kb   = qb + (size_t)Bc * Hc * Sc * DKc; // [B*H, S, DK]
  float* vb   = kb + (size_t)Bc * Hc * Sc * DKc; // [B*H, S, DV]
  float* cbuf = vb + (size_t)Bc * Hc * Sc * DVc; // [B, S, H*DV]

  // 1) QKV: one block per (bh, row tile) = 64*64 = 4096 blocks, 10 waves each
  qkv_kernel<<<dim3(4096), dim3(320), 0, stream>>>(
      x, Wq, bq, Wk, bk, Wv, bv, qb, kb, vb);

  // 2) scores: one block per (bh, row tile); 8 waves x 8 column tiles
  scores_kernel<<<dim3(4096), dim3(256), 0, stream>>>(qb, kb, wts);

  // 3) softmax over B*H*S = 65536 rows
  softmax_kernel<<<dim3(Bc * Hc * Sc), dim3(256), 0, stream>>>(wts);

  // 4) weights @ v: one block per (bh, 8 row tiles) = 64*8 = 512 blocks
  av_kernel<<<dim3(512), dim3(256), 0, stream>>>(wts, vb, cbuf);

  // 5) output projection: 4096 wave-strips / 8 = 512 blocks
  oproj_kernel<<<dim3(512), dim3(256), 0, stream>>>(cbuf, Wo, bo, out);
}